// AssocScan_40596030882228
// MI455X (gfx1250) — compile-verified
//
#include <hip/hip_runtime.h>
#include <stdint.h>

// Recurrence scan: out[b,s,d] = g[b,s,d]*out[b,s-1,d] + x[b,s,d], out[b,-1,d]=prev[b,d]
// B=4, S=4096, D=4096, fp32. Memory-bound: 768MB @ 23.3TB/s ~= 33us roofline.
// Single pass over HBM: TDM loads tiles to LDS, chained inter-chunk carries,
// async LDS->global stores.

#define Bsz 4
#define Ssz 4096
#define Dsz 4096

#define CH     128                    // channels per block (contiguous d within one b)
#define LCH    128                    // timesteps per chunk
#define NTHR   128                    // one thread per channel
#define NWAVE  (NTHR / 32)            // 4 wave32s
#define COLS   ((Bsz * Dsz) / CH)     // 128 channel columns
#define CHUNKS (Ssz / LCH)            // 32 chunks -> carry chain depth 32

#if __has_builtin(__builtin_amdgcn_global_load_async_to_lds_b128) && \
    __has_builtin(__builtin_amdgcn_global_store_async_from_lds_b128) && \
    __has_builtin(__builtin_amdgcn_s_wait_asynccnt)
#define USE_ASYNC 1
#else
#define USE_ASYNC 0
#endif

#if __has_builtin(__builtin_amdgcn_tensor_load_to_lds) && \
    __has_builtin(__builtin_amdgcn_s_wait_tensorcnt)
#define USE_TDM 1
#else
#define USE_TDM 0
#endif

typedef int          v4i __attribute__((ext_vector_type(4)));
typedef unsigned int v4u __attribute__((ext_vector_type(4)));
typedef int          v8i __attribute__((ext_vector_type(8)));
#define GAS __attribute__((address_space(1)))
#define LAS __attribute__((address_space(3)))

#if USE_ASYNC
#define ASYNC_WAIT0() __builtin_amdgcn_s_wait_asynccnt(0)
#else
#define ASYNC_WAIT0()
#endif

#if USE_TDM
// One D# descriptor: 2D tile, 128 rows x 128 f32 (512B), row stride Dsz*4 bytes,
// global -> LDS, no padding / iteration / multicast / completion-barrier.
__device__ __forceinline__ void tdm_load_tile(const float* gsrc, unsigned lds_off) {
  const uint64_t ga = (uint64_t)(uintptr_t)gsrc;
  v4u g0;
  g0.x = 1u;                                          // count=1 (valid), user mode
  g0.y = lds_off;                                     // lds_addr (bytes)
  g0.z = (unsigned)ga;                                // global_addr[31:0]
  g0.w = (unsigned)((ga >> 32) & 0x1FFFFFFu)          // global_addr[56:32]
         | (2u << 30);                                // type=2 ("image")
  v8i g1;
  g1[0] = 0x20000;                                    // data_size=2 (4B); mask=0
  g1[1] = (int)(0x1000u << 16);                       // tensor_dim0=4096 lo16 @ bits[63:48]
  g1[2] = (int)(0x1000u << 16);                       // dim0 hi=0; tensor_dim1=4096 lo16
  g1[3] = (int)(128u << 16);                          // dim1 hi=0; tile_dim0=128
  g1[4] = 128;                                        // tile_dim1=128; tile_dim2=0
  g1[5] = Dsz;                                        // tensor_dim0_stride=4096 (lo32)
  g1[6] = 0;                                          // stride hi; tensor_dim1_stride lo (unused, 2D)
  g1[7] = 0;
  const v4i gz4 = {0, 0, 0, 0};                       // groups 2/3: 2D tensor
  const v8i gz8 = {0, 0, 0, 0, 0, 0, 0, 0};
  __builtin_amdgcn_tensor_load_to_lds(g0, g1, gz4, gz4, gz8, 0);
}
#endif

__global__ __launch_bounds__(NTHR) void assoc_scan_chained(
    const float* __restrict__ g, const float* __restrict__ x,
    const float* __restrict__ prev, float* __restrict__ out,
    float* __restrict__ carry, int* __restrict__ flags) {
  // 128KB LDS: gates tile + inputs tile (reused as prefix-product / output tile)
  __shared__ __align__(16) float s_g[LCH * CH];
  __shared__ __align__(16) float s_x[LCH * CH];

  const int tid  = threadIdx.x;
  const int lane = tid & 31;
  const int wv   = tid >> 5;
  const int col   = blockIdx.x;      // channel column
  const int chunk = blockIdx.y;      // s-chunk; linear block id is chunk-major
  const int c0 = col * CH;           // first flat channel (c = b*D + d)
  const int b  = c0 / Dsz;           // CH divides D -> block stays in one batch
  const int d0 = c0 % Dsz;
  const int s0 = chunk * LCH;
  const size_t base = (size_t)b * Ssz * Dsz + (size_t)s0 * Dsz + (size_t)d0;

  // ---- Stage global tile into LDS ----
#if USE_TDM
  if (wv == 0) {  // TDM ignores EXEC; issue once per block, not once per wave
    const unsigned lofs_g = (unsigned)(uintptr_t)(LAS void*)&s_g[0];
    const unsigned lofs_x = (unsigned)(uintptr_t)(LAS void*)&s_x[0];
    tdm_load_tile(g + base, lofs_g);
    tdm_load_tile(x + base, lofs_x);
    __builtin_amdgcn_s_wait_tensorcnt(0);
  }
#else
  for (int r = wv; r < LCH; r += NWAVE) {
    const float* gp = g + base + (size_t)r * Dsz + (size_t)lane * 4;
    const float* xp = x + base + (size_t)r * Dsz + (size_t)lane * 4;
    float* lg = &s_g[r * CH + lane * 4];
    float* lx = &s_x[r * CH + lane * 4];
#if USE_ASYNC
    __builtin_amdgcn_global_load_async_to_lds_b128((GAS v4i*)gp, (LAS v4i*)lg, 0, 0);
    __builtin_amdgcn_global_load_async_to_lds_b128((GAS v4i*)xp, (LAS v4i*)lx, 0, 0);
#else
    *(float4*)lg = *(const float4*)gp;
    *(float4*)lx = *(const float4*)xp;
#endif
  }
  ASYNC_WAIT0();
#endif
  __syncthreads();

  // ---- Phase A: local scan from 0; keep running prefix y and cumprod P in LDS ----
  float P = 1.0f, y = 0.0f;
#pragma unroll 8
  for (int s = 0; s < LCH; ++s) {
    const float gv = s_g[s * CH + tid];
    const float xv = s_x[s * CH + tid];
    y = fmaf(gv, y, xv);
    P = P * gv;
    s_g[s * CH + tid] = P;   // cumprod of gates up to s
    s_x[s * CH + tid] = y;   // local scan value (zero initial state)
  }

  // ---- Carry: receive from predecessor chunk (or prev for chunk 0) ----
  const int c = c0 + tid;    // flat channel index == b*D + d
  float cin;
  if (chunk == 0) {
    cin = prev[c];
  } else {
    const int* fl = flags + (size_t)(chunk - 1) * COLS + col;
    while (__hip_atomic_load(fl, __ATOMIC_ACQUIRE, __HIP_MEMORY_SCOPE_AGENT) == 0) {
      __builtin_amdgcn_s_sleep(1);
    }
    cin = carry[(size_t)(chunk - 1) * (Bsz * Dsz) + c];
  }

  // ---- Publish carry_out = P*cin + y for the next chunk ----
  if (chunk + 1 < CHUNKS) {
    carry[(size_t)chunk * (Bsz * Dsz) + c] = fmaf(P, cin, y);
    __syncthreads();
    if (tid == 0) {
      __threadfence();
      __hip_atomic_store(flags + (size_t)chunk * COLS + col, 1,
                         __ATOMIC_RELEASE, __HIP_MEMORY_SCOPE_AGENT);
    }
  }

  // ---- Phase B: elementwise fixup out_s = P_s*cin + y_s, into LDS ----
#pragma unroll 8
  for (int s = 0; s < LCH; ++s) {
    s_x[s * CH + tid] = fmaf(s_g[s * CH + tid], cin, s_x[s * CH + tid]);
  }
  __syncthreads();

  // ---- Async store LDS tile -> out ----
  for (int r = wv; r < LCH; r += NWAVE) {
    float* op = out + base + (size_t)r * Dsz + (size_t)lane * 4;
    const float* lx = &s_x[r * CH + lane * 4];
#if USE_ASYNC
    __builtin_amdgcn_global_store_async_from_lds_b128((GAS v4i*)op, (LAS v4i*)lx, 0, 0);
#else
    *(float4*)op = *(const float4*)lx;
#endif
  }
  ASYNC_WAIT0();  // LDS must stay valid until async stores drain
}

extern "C" void kernel_launch(void* const* d_in, const int* in_sizes, int n_in,
                              void* d_out, int out_size, void* d_ws, size_t ws_size,
                              hipStream_t stream) {
  const float* g    = (const float*)d_in[0];  // gates  [B,S,D] f32
  const float* x    = (const float*)d_in[1];  // inputs [B,S,D] f32
  const float* prev = (const float*)d_in[2];  // prev   [B,1,D] f32
  float* out = (float*)d_out;

  // Workspace: 2MB of inter-chunk carries + 16KB of ready flags.
  float* carry = (float*)d_ws;
  int* flags = (int*)((char*)d_ws + sizeof(float) * (size_t)CHUNKS * Bsz * Dsz);

  // Flags must be zero at the start of every (replayed) call.
  (void)hipMemsetAsync(flags, 0, sizeof(int) * (size_t)CHUNKS * COLS, stream);

  dim3 grid(COLS, CHUNKS);  // linear block id = col + chunk*COLS: predecessors dispatch first
  dim3 blk(NTHR);
  hipLaunchKernelGGL(assoc_scan_chained, grid, blk, 0, stream, g, x, prev, out, carry, flags);
}